// HyperConnection_31516470018406
// MI455X (gfx1250) — compile-verified
//
#include <hip/hip_runtime.h>
#include <math.h>

typedef __attribute__((ext_vector_type(2))) float v2f;
typedef __attribute__((ext_vector_type(4))) float v4f;
typedef __attribute__((ext_vector_type(8))) float v8f;
typedef __attribute__((ext_vector_type(4))) int   v4i;

#if defined(__has_builtin)
#  if __has_builtin(__builtin_amdgcn_global_load_async_to_lds_b128) && \
      __has_builtin(__builtin_amdgcn_s_wait_asynccnt)
#    define USE_ASYNC_LDS 1
#  endif
#endif
#ifndef USE_ASYNC_LDS
#  define USE_ASYNC_LDS 0
#endif

#define D_DIM      2048
#define TOK_PER_WG 16
#define NTHREADS   256
#define NWAVES     8
#define ROW_STRIDE (D_DIM + 4)   /* 2052 dwords: row m starts at bank 4m -> conflict-free WMMA A loads */
#define RMS_EPS    1e-5f

__global__ __launch_bounds__(NTHREADS)
void HyperConnection_kernel(const float* __restrict__ h,
                            const float* __restrict__ h_o,
                            const float* __restrict__ st_alpha,   /* [4,5]  */
                            const float* __restrict__ st_beta,    /* [4]    */
                            const float* __restrict__ a_fn,       /* [D,5]  */
                            const float* __restrict__ a_scale,    /* [1]    */
                            const float* __restrict__ b_fn,       /* [D]    */
                            const float* __restrict__ b_scale,    /* [1]    */
                            const float* __restrict__ rmsw,       /* [D]    */
                            float* __restrict__ out)
{
    __shared__ float h_lds[TOK_PER_WG * ROW_STRIDE];        /* 131328 B */
    __shared__ float wsum[TOK_PER_WG][16];                  /* partial sumsq, reused for proj */
    __shared__ float rstd_lds[TOK_PER_WG];
    __shared__ float cbuf[NWAVES * 32 * 8];                 /* per-wave WMMA accumulators */
    __shared__ float coef[TOK_PER_WG][2];

    const int t = threadIdx.x;
    const long tile = (long)blockIdx.x * TOK_PER_WG;        /* first token of this tile */
    const float* hbase  = h   + tile * D_DIM;
    const float* hobase = h_o + tile * D_DIM;
    float*       obase  = out + tile * D_DIM;

    /* ---- Phase 0: stage 16 rows of h into LDS ----
     * Prefer the CDNA5 async copy path (GLOBAL_LOAD_ASYNC_TO_LDS_B128,
     * tracked by ASYNCcnt, no VGPR staging); fall back to b128 loads.   */
    for (int it = 0; it < 32; ++it) {
        int flat = it * (NTHREADS * 4) + t * 4;
        int row  = flat >> 11;            /* /2048 */
        int col  = flat & (D_DIM - 1);
#if USE_ASYNC_LDS
        __builtin_amdgcn_global_load_async_to_lds_b128(
            (v4i*)(hbase + row * D_DIM + col),
            (v4i*)&h_lds[row * ROW_STRIDE + col], 0, 0);
#else
        v4f v = *(const v4f*)(hbase + row * D_DIM + col);
        *(v4f*)(&h_lds[row * ROW_STRIDE + col]) = v;
#endif
    }
    /* warm L2 with this tile's h_o while we do RMS + WMMA (used in phase 4) */
    {
        const char* p = (const char*)hobase;
        #pragma unroll
        for (int q = 0; q < 4; ++q)
            __builtin_prefetch(p + (size_t)t * 512 + q * 128, 0, 1);
    }
#if USE_ASYNC_LDS
    __builtin_amdgcn_s_wait_asynccnt(0);
#endif
    __syncthreads();

    /* ---- Phase 1: per-row sum of squares -> rstd ---- */
    {
        int row = t >> 4, sub = t & 15;
        const float* r = &h_lds[row * ROW_STRIDE];
        float s = 0.0f;
        #pragma unroll 8
        for (int j = 0; j < 128; ++j) {   /* stride-16 access: conflict-free */
            float x = r[j * 16 + sub];
            s += x * x;
        }
        wsum[row][sub] = s;
    }
    __syncthreads();
    if (t < TOK_PER_WG) {
        float s = 0.0f;
        #pragma unroll
        for (int j = 0; j < 16; ++j) s += wsum[t][j];
        rstd_lds[t] = rsqrtf(s * (1.0f / (float)D_DIM) + RMS_EPS);
    }
    __syncthreads();

    /* ---- Phase 2: projection via V_WMMA_F32_16X16X4_F32 ----
     * proj[m,n] = rstd[m] * sum_k h[m,k] * rmsw[k] * W[k,n]
     * A (16x4 f32): raw h from LDS (lane l&15 = M; VGPR0/1 = K 2*(l>>4)+{0,1}).
     * B (4x16 f32): rmsw[k] * [a_fn | b_fn | 0..] built per lane, branch-free.
     * rstd[m] deferred to the phase-3 reduction (linear).
     * K=2048 split across 8 waves, 64 WMMAs each.                         */
    const int wave = t >> 5, lane = t & 31;
    const int m  = lane & 15;
    const int hi = lane >> 4;
    const int n  = m;                                    /* B column for this lane */

    /* branch-free B column source: keeps EXEC all-ones around WMMA */
    const float* wcol  = (n < 5) ? (a_fn + n) : b_fn;
    const int    wstr  = (n < 5) ? 5 : 1;
    const float  wmask = (n <= 5) ? 1.0f : 0.0f;

    v8f c = {};
    const int kbeg = wave * (D_DIM / NWAVES);
    const float* arow = &h_lds[m * ROW_STRIDE];
    for (int k0 = kbeg; k0 < kbeg + D_DIM / NWAVES; k0 += 4) {
        const int kk = k0 + 2 * hi;
        v2f a, b;
        v2f rw = *(const v2f*)(rmsw + kk);               /* kk even -> 8B aligned */
        a.x = arow[kk];                                  /* raw ds_load_2addr_b64 */
        a.y = arow[kk + 1];
        b.x = wcol[kk * wstr]       * (rw.x * wmask);
        b.y = wcol[(kk + 1) * wstr] * (rw.y * wmask);
        c = __builtin_amdgcn_wmma_f32_16x16x4_f32(false, a, false, b,
                                                  (short)0, c, false, false);
    }
    #pragma unroll
    for (int j = 0; j < 8; ++j) cbuf[(wave * 32 + lane) * 8 + j] = c[j];
    __syncthreads();

    /* ---- Phase 3: cross-wave reduce of C (x rstd), then coefficients ---- */
    if (t < 96) {                                        /* 16 tokens x 6 cols */
        int mm = t / 6, nn = t % 6;
        int vg = mm & 7, li = nn + 16 * (mm >> 3);       /* C/D VGPR layout */
        float s = 0.0f;
        #pragma unroll
        for (int w = 0; w < NWAVES; ++w) s += cbuf[(w * 32 + li) * 8 + vg];
        wsum[mm][nn] = s * rstd_lds[mm];                 /* proj[token][col] */
    }
    __syncthreads();
    if (t < TOK_PER_WG) {
        const float asc = a_scale[0], bsc = b_scale[0];
        float A = 0.0f;
        #pragma unroll
        for (int cc = 1; cc <= 4; ++cc) {
            float off    = tanhf(wsum[t][cc]) * asc;
            float colsum = st_alpha[cc] + st_alpha[5 + cc] +
                           st_alpha[10 + cc] + st_alpha[15 + cc];
            A += 4.0f * off + colsum;
        }
        float Bc = 4.0f * tanhf(wsum[t][5]) * bsc +
                   (st_beta[0] + st_beta[1] + st_beta[2] + st_beta[3]);
        coef[t][0] = A;
        coef[t][1] = Bc;
    }
    __syncthreads();

    /* ---- Phase 4: stream out = A*h(LDS) + B*h_o(global), non-temporal b128 ---- */
    for (int it = 0; it < 32; ++it) {
        int flat = it * (NTHREADS * 4) + t * 4;
        int row  = flat >> 11;
        int col  = flat & (D_DIM - 1);
        float A = coef[row][0], Bc = coef[row][1];
        v4f hv = *(const v4f*)(&h_lds[row * ROW_STRIDE + col]);
        v4f ho = __builtin_nontemporal_load((const v4f*)(hobase + row * D_DIM + col));
        v4f o;
        o.x = A * hv.x + Bc * ho.x;
        o.y = A * hv.y + Bc * ho.y;
        o.z = A * hv.z + Bc * ho.z;
        o.w = A * hv.w + Bc * ho.w;
        __builtin_nontemporal_store(o, (v4f*)(obase + row * D_DIM + col));
    }
}

extern "C" void kernel_launch(void* const* d_in, const int* in_sizes, int n_in,
                              void* d_out, int out_size, void* d_ws, size_t ws_size,
                              hipStream_t stream) {
    (void)n_in; (void)d_ws; (void)ws_size; (void)out_size;
    const float* h        = (const float*)d_in[0];
    const float* h_o      = (const float*)d_in[1];
    const float* st_alpha = (const float*)d_in[2];
    const float* st_beta  = (const float*)d_in[3];
    const float* a_fn     = (const float*)d_in[4];
    const float* a_scale  = (const float*)d_in[5];
    const float* b_fn     = (const float*)d_in[6];
    const float* b_scale  = (const float*)d_in[7];
    const float* rmsw     = (const float*)d_in[8];
    float* out = (float*)d_out;

    const int n_tokens = in_sizes[0] / D_DIM;            /* B*L = 16384 */
    const int n_blocks = n_tokens / TOK_PER_WG;          /* 1024 */
    HyperConnection_kernel<<<n_blocks, NTHREADS, 0, stream>>>(
        h, h_o, st_alpha, st_beta, a_fn, a_scale, b_fn, b_scale, rmsw, out);
}